// MKG_UE_49323404427362
// MI455X (gfx1250) — compile-verified
//
#include <hip/hip_runtime.h>
#include <math.h>

// ---- static problem config (matches reference) ----
#define N_USERS 40000
#define N_ITEMS 30000
#define N_ENT   80000
#define DIM     64
#define E_KG    800000
#define E_UI    600000

typedef __attribute__((ext_vector_type(2))) float v2f;
typedef __attribute__((ext_vector_type(8))) float v8f;

#define RSQRT_DK 0.17677669529663687f   // 1/sqrt(32)

// device-scope relaxed atomics -> native global_atomic_add_f32 / max_u32
static __device__ __forceinline__ void atomAddF(float* p, float v) {
    __hip_atomic_fetch_add(p, v, __ATOMIC_RELAXED, __HIP_MEMORY_SCOPE_AGENT);
}
static __device__ __forceinline__ void atomMaxU(unsigned* p, unsigned v) {
    __hip_atomic_fetch_max(p, v, __ATOMIC_RELAXED, __HIP_MEMORY_SCOPE_AGENT);
}

// ---- float <-> order-preserving uint key (for atomicMax on floats) ----
static __device__ __forceinline__ unsigned fkey(float f) {
    unsigned u = __float_as_uint(f);
    return (u & 0x80000000u) ? ~u : (u | 0x80000000u);
}
static __device__ __forceinline__ float funkey(unsigned k) {
    unsigned u = (k & 0x80000000u) ? (k & 0x7fffffffu) : ~k;
    return __uint_as_float(u);
}

// ---------------------------------------------------------------------------
// fill
// ---------------------------------------------------------------------------
__global__ void k_fill_f32(float* p, long n, float v) {
    long i  = (long)blockIdx.x * blockDim.x + threadIdx.x;
    long st = (long)gridDim.x * blockDim.x;
    for (; i < n; i += st) p[i] = v;
}

// ---------------------------------------------------------------------------
// Q = src[nrows,64] @ W[64,64], fp32 WMMA (v_wmma_f32_16x16x4_f32).
// One wave -> one 16x64 output tile. nrows must be a multiple of 16.
// A (16x4 f32): lanes 0-15 hold K={0,1}, lanes 16-31 hold K={2,3}.
// B (4x16 f32): mirrored (lanes 0-15 N, VGPR0/1 = K pair).
// C/D (16x16 f32): VGPR r: lanes 0-15 -> M=r, lanes 16-31 -> M=r+8.
// ---------------------------------------------------------------------------
__global__ void k_project_wmma(const float* __restrict__ src,
                               const float* __restrict__ W,
                               float* __restrict__ dst, int nrows) {
    int wave    = (blockIdx.x * blockDim.x + threadIdx.x) >> 5;
    int lane    = threadIdx.x & 31;
    int rowBase = wave * 16;
    if (rowBase >= nrows) return;

    int lm    = lane & 15;
    int khalf = (lane >> 4) * 2;

    v8f acc0 = {}, acc1 = {}, acc2 = {}, acc3 = {};
    const float* arow = src + (long)(rowBase + lm) * DIM;

#pragma unroll
    for (int kk = 0; kk < 16; ++kk) {
        int k = kk * 4 + khalf;
        v2f a = *(const v2f*)(arow + k);
        const float* w0 = W + (long)k * DIM;
        const float* w1 = w0 + DIM;
        v2f b0 = { w0[lm +  0], w1[lm +  0] };
        v2f b1 = { w0[lm + 16], w1[lm + 16] };
        v2f b2 = { w0[lm + 32], w1[lm + 32] };
        v2f b3 = { w0[lm + 48], w1[lm + 48] };
        acc0 = __builtin_amdgcn_wmma_f32_16x16x4_f32(false, a, false, b0, (short)0, acc0, false, false);
        acc1 = __builtin_amdgcn_wmma_f32_16x16x4_f32(false, a, false, b1, (short)0, acc1, false, false);
        acc2 = __builtin_amdgcn_wmma_f32_16x16x4_f32(false, a, false, b2, (short)0, acc2, false, false);
        acc3 = __builtin_amdgcn_wmma_f32_16x16x4_f32(false, a, false, b3, (short)0, acc3, false, false);
    }

    int   mOff = (lane >> 4) * 8;
    float* drow = dst + (long)rowBase * DIM;
#pragma unroll
    for (int r = 0; r < 8; ++r) {
        long m = mOff + r;
        drow[m * DIM + lm +  0] = acc0[r];
        drow[m * DIM + lm + 16] = acc1[r];
        drow[m * DIM + lm + 32] = acc2[r];
        drow[m * DIM + lm + 48] = acc3[r];
    }
}

// ---------------------------------------------------------------------------
// per-edge 2-head attention logits: attn[e,h] = sum_d q*k*rel / sqrt(32)
// wave per edge; lane l owns dims {l (head0), l+32 (head1)}
// ---------------------------------------------------------------------------
__global__ void k_edge_attn(const float* __restrict__ Q,
                            const float* __restrict__ rel,
                            const int* __restrict__ head,
                            const int* __restrict__ tail,
                            const int* __restrict__ etype,
                            float* __restrict__ attn) {
    int wv   = (blockIdx.x * blockDim.x + threadIdx.x) >> 5;
    int lane = threadIdx.x & 31;
    int nw   = (gridDim.x * blockDim.x) >> 5;
    for (int e = wv; e < E_KG; e += nw) {
        long h = head[e], t = tail[e];
        long r = etype[e] - 1;
        float p0 = Q[h * DIM + lane]      * Q[t * DIM + lane]      * rel[r * DIM + lane];
        float p1 = Q[h * DIM + 32 + lane] * Q[t * DIM + 32 + lane] * rel[r * DIM + 32 + lane];
#pragma unroll
        for (int m = 16; m; m >>= 1) {
            p0 += __shfl_xor(p0, m, 32);
            p1 += __shfl_xor(p1, m, 32);
        }
        if (lane == 0) {
            attn[2L * e]     = p0 * RSQRT_DK;
            attn[2L * e + 1] = p1 * RSQRT_DK;
        }
    }
}

// ---------------------------------------------------------------------------
// phase A: segment max of mean-logit, degree count, item_attn_sumed
// ---------------------------------------------------------------------------
__global__ void k_phaseA_stats(const float* __restrict__ attn,
                               const int* __restrict__ head,
                               const int* __restrict__ tail,
                               unsigned* __restrict__ segMaxA,
                               float* __restrict__ deg,
                               float* __restrict__ itemOut) {
    int e = blockIdx.x * blockDim.x + threadIdx.x;
    if (e >= E_KG) return;
    float l = 0.5f * (attn[2L * e] + attn[2L * e + 1]);
    int h = head[e], t = tail[e];
    atomMaxU(&segMaxA[h], fkey(l));
    atomAddF(&deg[h], 1.0f);
    if (h < N_ITEMS) atomAddF(&itemOut[h], l);
    if (t < N_ITEMS) atomAddF(&itemOut[t], l);
}

__global__ void k_phaseA_expsum(const float* __restrict__ attn,
                                const int* __restrict__ head,
                                const unsigned* __restrict__ segMaxA,
                                float* __restrict__ segSumA) {
    int e = blockIdx.x * blockDim.x + threadIdx.x;
    if (e >= E_KG) return;
    float l = 0.5f * (attn[2L * e] + attn[2L * e + 1]);
    int h = head[e];
    atomAddF(&segSumA[h], expf(l - funkey(segMaxA[h])));
}

__global__ void k_phaseA_score(const float* __restrict__ attn,
                               const int* __restrict__ head,
                               const unsigned* __restrict__ segMaxA,
                               const float* __restrict__ segSumA,
                               const float* __restrict__ deg,
                               float* __restrict__ outScore) {
    int e = blockIdx.x * blockDim.x + threadIdx.x;
    if (e >= E_KG) return;
    float l = 0.5f * (attn[2L * e] + attn[2L * e + 1]);
    int h = head[e];
    float ex = expf(l - funkey(segMaxA[h]));
    outScore[e] = ex / (segSumA[h] + 1e-16f) * deg[h];
}

// ---------------------------------------------------------------------------
// hop softmax (per head) passes
// ---------------------------------------------------------------------------
__global__ void k_hop_max(const float* __restrict__ attn,
                          const int* __restrict__ head,
                          unsigned* __restrict__ segMaxH) {
    int e = blockIdx.x * blockDim.x + threadIdx.x;
    if (e >= E_KG) return;
    long h = head[e];
    atomMaxU(&segMaxH[2 * h],     fkey(attn[2L * e]));
    atomMaxU(&segMaxH[2 * h + 1], fkey(attn[2L * e + 1]));
}

__global__ void k_hop_expsum(const float* __restrict__ attn,
                             const int* __restrict__ head,
                             const unsigned* __restrict__ segMaxH,
                             float* __restrict__ segSumH,
                             float* __restrict__ edgeEx) {
    int e = blockIdx.x * blockDim.x + threadIdx.x;
    if (e >= E_KG) return;
    long h = head[e];
    float e0 = expf(attn[2L * e]     - funkey(segMaxH[2 * h]));
    float e1 = expf(attn[2L * e + 1] - funkey(segMaxH[2 * h + 1]));
    edgeEx[2L * e]     = e0;
    edgeEx[2L * e + 1] = e1;
    atomAddF(&segSumH[2 * h],     e0);
    atomAddF(&segSumH[2 * h + 1], e1);
}

// ---------------------------------------------------------------------------
// KG aggregation: eAgg[head] += (e[tail]*rel) * softmax(head)  (wave/edge)
// ---------------------------------------------------------------------------
__global__ void k_kg_agg(const float* __restrict__ eSrc,
                         const float* __restrict__ rel,
                         const float* __restrict__ edgeEx,
                         const float* __restrict__ segSumH,
                         const int* __restrict__ head,
                         const int* __restrict__ tail,
                         const int* __restrict__ etype,
                         float* __restrict__ eAgg) {
    int wv   = (blockIdx.x * blockDim.x + threadIdx.x) >> 5;
    int lane = threadIdx.x & 31;
    int nw   = (gridDim.x * blockDim.x) >> 5;
    for (int e = wv; e < E_KG; e += nw) {
        long h = head[e], t = tail[e];
        long r = etype[e] - 1;
        float s0 = edgeEx[2L * e]     / (segSumH[2 * h]     + 1e-16f);
        float s1 = edgeEx[2L * e + 1] / (segSumH[2 * h + 1] + 1e-16f);
        float v0 = eSrc[t * DIM + lane]      * rel[r * DIM + lane]      * s0;
        float v1 = eSrc[t * DIM + 32 + lane] * rel[r * DIM + 32 + lane] * s1;
        atomAddF(&eAgg[h * DIM + lane],      v0);
        atomAddF(&eAgg[h * DIM + 32 + lane], v1);
    }
}

// ---------------------------------------------------------------------------
// bipartite user<->item weighted aggregation (wave/edge)
// ---------------------------------------------------------------------------
__global__ void k_bipartite(const float* __restrict__ eSrc,
                            const float* __restrict__ uSrc,
                            const float* __restrict__ w,
                            const int* __restrict__ iu,
                            const int* __restrict__ ii,
                            float* __restrict__ uAgg,
                            float* __restrict__ eAgg) {
    int wv   = (blockIdx.x * blockDim.x + threadIdx.x) >> 5;
    int lane = threadIdx.x & 31;
    int nw   = (gridDim.x * blockDim.x) >> 5;
    for (int e = wv; e < E_UI; e += nw) {
        long u = iu[e], it = ii[e];
        float ww = w[e];
        atomAddF(&uAgg[u * DIM + lane],       ww * eSrc[it * DIM + lane]);
        atomAddF(&uAgg[u * DIM + 32 + lane],  ww * eSrc[it * DIM + 32 + lane]);
        atomAddF(&eAgg[it * DIM + lane],      ww * uSrc[u * DIM + lane]);
        atomAddF(&eAgg[it * DIM + 32 + lane], ww * uSrc[u * DIM + 32 + lane]);
    }
}

// ---------------------------------------------------------------------------
// l2-normalize rows of agg -> cur, and res += cur   (wave/row)
// ---------------------------------------------------------------------------
__global__ void k_norm_res(const float* __restrict__ agg,
                           float* __restrict__ cur,
                           float* __restrict__ res, int nrows) {
    int wv   = (blockIdx.x * blockDim.x + threadIdx.x) >> 5;
    int lane = threadIdx.x & 31;
    int nw   = (gridDim.x * blockDim.x) >> 5;
    for (int row = wv; row < nrows; row += nw) {
        float x0 = agg[(long)row * DIM + lane];
        float x1 = agg[(long)row * DIM + 32 + lane];
        float ss = x0 * x0 + x1 * x1;
#pragma unroll
        for (int m = 16; m; m >>= 1) ss += __shfl_xor(ss, m, 32);
        float inv = 1.0f / fmaxf(sqrtf(ss), 1e-12f);
        float y0 = x0 * inv, y1 = x1 * inv;
        cur[(long)row * DIM + lane]      = y0;
        cur[(long)row * DIM + 32 + lane] = y1;
        res[(long)row * DIM + lane]      += y0;
        res[(long)row * DIM + 32 + lane] += y1;
    }
}

// ---------------------------------------------------------------------------
extern "C" void kernel_launch(void* const* d_in, const int* in_sizes, int n_in,
                              void* d_out, int out_size, void* d_ws, size_t ws_size,
                              hipStream_t stream) {
    const float* user_emb   = (const float*)d_in[0];
    const float* entity_emb = (const float*)d_in[1];
    const float* rel        = (const float*)d_in[2];
    const float* W_Q        = (const float*)d_in[3];
    const float* inter_w    = (const float*)d_in[4];
    const int*   head       = (const int*)d_in[5];
    const int*   tail       = (const int*)d_in[6];
    const int*   etype      = (const int*)d_in[7];
    const int*   iu         = (const int*)d_in[8];
    const int*   ii         = (const int*)d_in[9];

    // output layout: [u_res 40000*64][e_res 80000*64][edge_attn 800000][item_attn 30000]
    float* out       = (float*)d_out;
    float* uRes      = out;
    float* eRes      = out + (long)N_USERS * DIM;
    float* edgeScore = out + (long)(N_USERS + N_ENT) * DIM;
    float* itemOut   = edgeScore + E_KG;

    // workspace layout (~97 MB of f32)
    float* ws = (float*)d_ws;
    float*    Q       = ws; ws += (long)N_ENT * DIM;
    float*    eCur    = ws; ws += (long)N_ENT * DIM;
    float*    uCur    = ws; ws += (long)N_USERS * DIM;
    float*    eAgg    = ws; ws += (long)N_ENT * DIM;
    float*    uAgg    = ws; ws += (long)N_USERS * DIM;
    float*    attn    = ws; ws += 2L * E_KG;
    float*    edgeEx  = ws; ws += 2L * E_KG;
    unsigned* segMaxA = (unsigned*)ws; ws += N_ENT;
    float*    segSumA = ws; ws += N_ENT;
    float*    deg     = ws; ws += N_ENT;
    unsigned* segMaxH = (unsigned*)ws; ws += 2L * N_ENT;
    float*    segSumH = ws; ws += 2L * N_ENT;

    auto fill0 = [&](float* p, long n) {
        k_fill_f32<<<(unsigned)((n + 255) / 256), 256, 0, stream>>>(p, n, 0.0f);
    };

    // residual init + accumulator zeroing
    hipMemcpyAsync(uRes, user_emb,   sizeof(float) * (long)N_USERS * DIM, hipMemcpyDeviceToDevice, stream);
    hipMemcpyAsync(eRes, entity_emb, sizeof(float) * (long)N_ENT   * DIM, hipMemcpyDeviceToDevice, stream);
    fill0(itemOut, N_ITEMS);
    fill0((float*)segMaxA, N_ENT);   // key 0 == -inf
    fill0(segSumA, N_ENT);
    fill0(deg, N_ENT);

    const int wmmaBlocks     = (N_ENT / 16 + 7) / 8;  // 8 waves/block
    const int edgeWaveBlocks = (E_KG + 7) / 8;
    const int edgeThBlocks   = (E_KG + 255) / 256;
    const int uiWaveBlocks   = (E_UI + 7) / 8;

    // Q projection + hop-0 attention (shared with phase A: logits = mean over heads)
    k_project_wmma<<<wmmaBlocks, 256, 0, stream>>>(entity_emb, W_Q, Q, N_ENT);
    k_edge_attn<<<edgeWaveBlocks, 256, 0, stream>>>(Q, rel, head, tail, etype, attn);

    // phase A: edge_attn_score + item_attn_sumed
    k_phaseA_stats <<<edgeThBlocks, 256, 0, stream>>>(attn, head, tail, segMaxA, deg, itemOut);
    k_phaseA_expsum<<<edgeThBlocks, 256, 0, stream>>>(attn, head, segMaxA, segSumA);
    k_phaseA_score <<<edgeThBlocks, 256, 0, stream>>>(attn, head, segMaxA, segSumA, deg, edgeScore);

    // two message-passing hops
    const float* eSrc = entity_emb;
    const float* uSrc = user_emb;
    for (int hop = 0; hop < 2; ++hop) {
        if (hop) {
            k_project_wmma<<<wmmaBlocks, 256, 0, stream>>>(eSrc, W_Q, Q, N_ENT);
            k_edge_attn<<<edgeWaveBlocks, 256, 0, stream>>>(Q, rel, head, tail, etype, attn);
        }
        fill0((float*)segMaxH, 2L * N_ENT);
        fill0(segSumH, 2L * N_ENT);
        fill0(eAgg, (long)N_ENT * DIM);
        fill0(uAgg, (long)N_USERS * DIM);

        k_hop_max   <<<edgeThBlocks, 256, 0, stream>>>(attn, head, segMaxH);
        k_hop_expsum<<<edgeThBlocks, 256, 0, stream>>>(attn, head, segMaxH, segSumH, edgeEx);
        k_kg_agg    <<<edgeWaveBlocks, 256, 0, stream>>>(eSrc, rel, edgeEx, segSumH,
                                                         head, tail, etype, eAgg);
        k_bipartite <<<uiWaveBlocks, 256, 0, stream>>>(eSrc, uSrc, inter_w, iu, ii, uAgg, eAgg);

        k_norm_res<<<(N_ENT   + 7) / 8, 256, 0, stream>>>(eAgg, eCur, eRes, N_ENT);
        k_norm_res<<<(N_USERS + 7) / 8, 256, 0, stream>>>(uAgg, uCur, uRes, N_USERS);

        eSrc = eCur;
        uSrc = uCur;
    }
}